// YoloLoss_4458176053314
// MI455X (gfx1250) — compile-verified
//
#include <hip/hip_runtime.h>
#include <hip/hip_bf16.h>
#include <math.h>

#define SGRID 14
#define L_COORD 5.0f
#define L_NOOBJ 0.5f
#define TILE 256

typedef float v2f __attribute__((ext_vector_type(2)));
typedef float v8f __attribute__((ext_vector_type(8)));

// Async staging path: only if the gfx1250 async-to-LDS builtins exist in this
// toolchain; otherwise fall back to direct loads (known-good round-1 path).
#if __has_builtin(__builtin_amdgcn_global_load_async_to_lds_b64) && \
    __has_builtin(__builtin_amdgcn_s_wait_asynccnt)
#define YOLO_ASYNC_STAGE 1
#else
#define YOLO_ASYNC_STAGE 0
#endif

#if YOLO_ASYNC_STAGE
// Exact parameter types per hipcc diagnostic: the builtin wants pointers to a
// 2 x int vector, global source in AS(1), LDS destination in AS(3).
typedef int b64vec __attribute__((vector_size(2 * sizeof(int))));
typedef __attribute__((address_space(1))) b64vec* global_b64_ptr;
typedef __attribute__((address_space(3))) b64vec* lds_b64_ptr;
#endif

// ---------------------------------------------------------------------------
// Block-wide reduction of four per-lane f32 partials with one
// V_WMMA_F32_16X16X4_F32 (A = ones -> D rows are column sums of B).
//  1. shfl_xor 16/8/4: lanes 0..3 hold 4 "class" partials per component.
//  2. lanes 0..3 of each wave -> lds[wave*16 + comp*4 + class]
//  3. all threads rebuild B (4x16 f32): column n = comp*4 + wavepair,
//     row k = class, folding wave w with w+4.
//     B VGPR0: lanes 0-15 = K=0, lanes 16-31 = K=2; VGPR1: K=1/K=3.
//  4. WMMA, then shfl_xor 1/2: lanes {4c..4c+3} hold the total of comp c.
// Requires blockDim==256, or 32 with lds[16..127] pre-zeroed. EXEC all ones.
// ---------------------------------------------------------------------------
__device__ __forceinline__ float block_reduce4_wmma(float r0, float r1,
                                                    float r2, float r3,
                                                    float* lds) {
  const int t = threadIdx.x;
  const int lane = t & 31;
  const int wave = t >> 5;

  r0 += __shfl_xor(r0, 16, 32);
  r1 += __shfl_xor(r1, 16, 32);
  r2 += __shfl_xor(r2, 16, 32);
  r3 += __shfl_xor(r3, 16, 32);
  r0 += __shfl_xor(r0, 8, 32);
  r1 += __shfl_xor(r1, 8, 32);
  r2 += __shfl_xor(r2, 8, 32);
  r3 += __shfl_xor(r3, 8, 32);
  r0 += __shfl_xor(r0, 4, 32);
  r1 += __shfl_xor(r1, 4, 32);
  r2 += __shfl_xor(r2, 4, 32);
  r3 += __shfl_xor(r3, 4, 32);

  if (lane < 4) {
    lds[wave * 16 + 0 * 4 + lane] = r0;
    lds[wave * 16 + 1 * 4 + lane] = r1;
    lds[wave * 16 + 2 * 4 + lane] = r2;
    lds[wave * 16 + 3 * 4 + lane] = r3;
  }
  __syncthreads();

  const int n = lane & 15;
  const int comp = n >> 2;
  const int wp = n & 3;
  const int k0 = (lane < 16) ? 0 : 2;
  float b0v = lds[wp * 16 + comp * 4 + k0] +
              lds[(wp + 4) * 16 + comp * 4 + k0];
  float b1v = lds[wp * 16 + comp * 4 + k0 + 1] +
              lds[(wp + 4) * 16 + comp * 4 + k0 + 1];

  v2f a;
  a[0] = 1.0f;
  a[1] = 1.0f;
  v2f b;
  b[0] = b0v;
  b[1] = b1v;
  v8f cacc = {};
  v8f dmat = __builtin_amdgcn_wmma_f32_16x16x4_f32(
      false, a, false, b, (short)0, cacc, false, false);

  float x = dmat[0];
  x += __shfl_xor(x, 1, 32);
  x += __shfl_xor(x, 2, 32);
  return x;  // lanes {4c..4c+3}: total of component c
}

// ---------------------------------------------------------------------------
// Per-cell YOLO loss terms. p = 30 pred floats, tc = 20 class floats.
// comp0 = reg, comp1 = contain, comp2 = noobj, comp3 = cls.
// ---------------------------------------------------------------------------
__device__ __forceinline__ void cell_loss(const float* p, const float* tc,
                                          float4 tb, float obj, float& r_reg,
                                          float& r_con, float& r_no,
                                          float& r_cls) {
  constexpr float kInvS = 1.0f / (float)SGRID;

  float cls_a = 0.0f;
#pragma unroll
  for (int c = 0; c < 20; ++c) {
    float d = p[10 + c] - tc[c];
    cls_a = fmaf(d, d, cls_a);
  }
  r_cls += obj * cls_a;

  r_no += L_NOOBJ * (1.0f - obj) * (p[4] * p[4] + p[9] * p[9]);

  float tcx = tb.x * kInvS, tcy = tb.y * kInvS;
  float tx1 = tcx - 0.5f * tb.z, ty1 = tcy - 0.5f * tb.w;
  float tx2 = tcx + 0.5f * tb.z, ty2 = tcy + 0.5f * tb.w;
  float tarea = (tx2 - tx1) * (ty2 - ty1);

  auto iou_vs_tgt = [&](float x, float y, float w, float h) -> float {
    float cx = x * kInvS, cy = y * kInvS;
    float x1 = cx - 0.5f * w, y1 = cy - 0.5f * h;
    float x2 = cx + 0.5f * w, y2 = cy + 0.5f * h;
    float lx = fmaxf(x1, tx1), ly = fmaxf(y1, ty1);
    float rx = fminf(x2, tx2), ry = fminf(y2, ty2);
    float iw = fmaxf(rx - lx, 0.0f), ih = fmaxf(ry - ly, 0.0f);
    float inter = iw * ih;
    float a1 = (x2 - x1) * (y2 - y1);
    return inter / (a1 + tarea - inter);
  };

  float iou1 = iou_vs_tgt(p[0], p[1], p[2], p[3]);
  float iou2 = iou_vs_tgt(p[5], p[6], p[7], p[8]);
  bool take1 = iou1 > iou2;
  float biou = take1 ? iou1 : iou2;
  float bx = take1 ? p[0] : p[5];
  float by = take1 ? p[1] : p[6];
  float bw = take1 ? p[2] : p[7];
  float bh = take1 ? p[3] : p[8];
  float bc = take1 ? p[4] : p[9];

  float dx = bx - tb.x, dy = by - tb.y;
  float sw = sqrtf(bw) - sqrtf(tb.z);
  float sh = sqrtf(bh) - sqrtf(tb.w);
  r_reg += L_COORD * obj * (dx * dx + dy * dy + sw * sw + sh * sh);

  float dc = bc - biou;
  r_con += obj * dc * dc;
}

// Direct (unstaged) processing of a single cell: b64 loads of pred,
// b128 loads of tcls/tbox.
__device__ __forceinline__ void cell_direct(const float* __restrict__ pred,
                                            const float* __restrict__ tbox,
                                            const float* __restrict__ tcls,
                                            const unsigned char* __restrict__
                                                objmap,
                                            int cell, float& r_reg,
                                            float& r_con, float& r_no,
                                            float& r_cls) {
  float p[30];
  const float2* pp = reinterpret_cast<const float2*>(pred + (size_t)cell * 30);
#pragma unroll
  for (int q = 0; q < 15; ++q) {
    float2 v = pp[q];
    p[2 * q] = v.x;
    p[2 * q + 1] = v.y;
  }
  float tc[20];
  const float4* pc = reinterpret_cast<const float4*>(tcls + (size_t)cell * 20);
#pragma unroll
  for (int q = 0; q < 5; ++q) {
    float4 v = pc[q];
    tc[4 * q] = v.x;
    tc[4 * q + 1] = v.y;
    tc[4 * q + 2] = v.z;
    tc[4 * q + 3] = v.w;
  }
  float4 tb = reinterpret_cast<const float4*>(tbox)[cell];
  float obj = objmap[cell] ? 1.0f : 0.0f;
  cell_loss(p, tc, tb, obj, r_reg, r_con, r_no, r_cls);
}

// ---------------------------------------------------------------------------
// Main pass. Staged variant: pred tiles (256 cells x 120B = 30720B) are
// streamed into LDS with coalesced global_load_async_to_lds_b64, double
// buffered, synchronized with s_wait_asynccnt (per-wave in-order retire:
// wait<=15 means the older tile's 15 ops are done). Every thread issues
// exactly 15 async ops per tile -> per-wave ASYNCcnt accounting is uniform.
// ---------------------------------------------------------------------------
__global__ void __launch_bounds__(256) yolo_loss_main(
    const float* __restrict__ pred, const float* __restrict__ tbox,
    const float* __restrict__ tcls, const unsigned char* __restrict__ objmap,
    float4* __restrict__ ws, int cells) {
  __shared__ float lds_red[128];
  float r_reg = 0.0f, r_con = 0.0f, r_no = 0.0f, r_cls = 0.0f;
  const int tid = (int)threadIdx.x;

#if YOLO_ASYNC_STAGE
  __shared__ float sp[2][TILE * 30];  // 2 x 30720 B pred tiles
  const int T = cells / TILE;         // full tiles
  const int G = (int)gridDim.x;
  int nt = ((int)blockIdx.x < T) ? (T - (int)blockIdx.x + G - 1) / G : 0;

  auto issue = [&](int i) {
    const int tile = (int)blockIdx.x + i * G;
    const float* g = pred + (size_t)tile * TILE * 30;
    float* s = &sp[i & 1][0];
#pragma unroll
    for (int k = 0; k < 15; ++k) {
      const int f2 = k * TILE + tid;  // float2 index inside the tile
      __builtin_amdgcn_global_load_async_to_lds_b64(
          (global_b64_ptr)(g + 2 * f2), (lds_b64_ptr)(s + 2 * f2), 0, 0);
    }
  };

  if (nt > 0) issue(0);
  if (nt > 1) issue(1);

  for (int i = 0; i < nt; ++i) {
    const int tile = (int)blockIdx.x + i * G;
    if (i + 1 < nt) {
      // next tile still in flight; this tile's 15 per-wave ops retired
      __builtin_amdgcn_s_wait_asynccnt(15);
      // hint: pull next tile's class data toward the caches
      __builtin_prefetch(tcls + (size_t)(tile + G) * TILE * 20 + tid * 20, 0,
                         0);
    } else {
      __builtin_amdgcn_s_wait_asynccnt(0);
    }
    __syncthreads();  // everyone's async portions are visible

    const int cell = tile * TILE + tid;
    float p[30];
    const float* sc = &sp[i & 1][tid * 30];
#pragma unroll
    for (int j = 0; j < 30; ++j) p[j] = sc[j];

    float tc[20];
    const float4* pc =
        reinterpret_cast<const float4*>(tcls + (size_t)cell * 20);
#pragma unroll
    for (int q = 0; q < 5; ++q) {
      float4 v = pc[q];
      tc[4 * q] = v.x;
      tc[4 * q + 1] = v.y;
      tc[4 * q + 2] = v.z;
      tc[4 * q + 3] = v.w;
    }
    float4 tb = reinterpret_cast<const float4*>(tbox)[cell];
    float obj = objmap[cell] ? 1.0f : 0.0f;
    cell_loss(p, tc, tb, obj, r_reg, r_con, r_no, r_cls);

    __syncthreads();  // buffer i&1 free to be overwritten
    if (i + 2 < nt) issue(i + 2);
  }

  // Remainder cells (cells % 256), handled directly by block 0.
  const int rstart = T * TILE;
  if ((int)blockIdx.x == 0 && rstart + tid < cells) {
    cell_direct(pred, tbox, tcls, objmap, rstart + tid, r_reg, r_con, r_no,
                r_cls);
  }
#else
  // Fallback: direct grid-stride loads (round-1 path).
  const int stride = (int)(gridDim.x * blockDim.x);
  for (int cell = (int)(blockIdx.x * blockDim.x) + tid; cell < cells;
       cell += stride) {
    if (cell + stride < cells)
      __builtin_prefetch(pred + (size_t)(cell + stride) * 30, 0, 0);
    cell_direct(pred, tbox, tcls, objmap, cell, r_reg, r_con, r_no, r_cls);
  }
#endif

  float x = block_reduce4_wmma(r_reg, r_con, r_no, r_cls, lds_red);
  float c1 = __shfl(x, 4, 32);
  float c2 = __shfl(x, 8, 32);
  float c3 = __shfl(x, 12, 32);
  if (tid == 0) {
    float4 o;
    o.x = x;   // reg
    o.y = c1;  // contain
    o.z = c2;  // noobj
    o.w = c3;  // cls
    ws[blockIdx.x] = o;
  }
}

// ---------------------------------------------------------------------------
// Finalize: one wave reduces all block partials and writes the 5 outputs.
// ---------------------------------------------------------------------------
__global__ void __launch_bounds__(32) yolo_loss_finalize(
    const float4* __restrict__ ws, int nb, float* __restrict__ out,
    float inv_n) {
  __shared__ float lds[128];
#pragma unroll
  for (int i = 0; i < 4; ++i) lds[threadIdx.x + 32 * i] = 0.0f;

  float r0 = 0.0f, r1 = 0.0f, r2 = 0.0f, r3 = 0.0f;
  for (int i = (int)threadIdx.x; i < nb; i += 32) {
    float4 v = ws[i];
    r0 += v.x;
    r1 += v.y;
    r2 += v.z;
    r3 += v.w;
  }

  float x = block_reduce4_wmma(r0, r1, r2, r3, lds);
  float reg = __shfl(x, 0, 32);
  float con = __shfl(x, 4, 32);
  float no = __shfl(x, 8, 32);
  float cls = __shfl(x, 12, 32);
  if (threadIdx.x == 0) {
    float total = reg + con + no + cls;
    out[0] = total * inv_n;
    out[1] = reg * inv_n;
    out[2] = con * inv_n;
    out[3] = no * inv_n;
    out[4] = cls * inv_n;
  }
}

extern "C" void kernel_launch(void* const* d_in, const int* in_sizes, int n_in,
                              void* d_out, int out_size, void* d_ws,
                              size_t ws_size, hipStream_t stream) {
  const float* pred = (const float*)d_in[0];          // (N,14,14,30) f32
  const float* tbox = (const float*)d_in[1];          // (N,14,14,4)  f32
  const float* tcls = (const float*)d_in[2];          // (N,14,14,20) f32
  const unsigned char* obj = (const unsigned char*)d_in[3];  // (N,14,14) bool
  float* out = (float*)d_out;                         // 5 f32

  const int cells = in_sizes[3];              // N * 14 * 14
  const int n_img = cells / (SGRID * SGRID);  // N
  const int tiles = cells / TILE;             // 3136 for the given shape

  int nblocks = tiles / 4;  // ~4 tiles per block, 784 blocks
  if (nblocks < 1) nblocks = 1;

  float4* ws = (float4*)d_ws;  // nblocks float4 partials

  yolo_loss_main<<<nblocks, 256, 0, stream>>>(pred, tbox, tcls, obj, ws,
                                              cells);
  yolo_loss_finalize<<<1, 32, 0, stream>>>(ws, nblocks, out,
                                           1.0f / (float)n_img);
}